// ResidualVQ_42468636622990
// MI455X (gfx1250) — compile-verified
//
#include <hip/hip_runtime.h>

typedef __attribute__((ext_vector_type(2))) float v2f;
typedef __attribute__((ext_vector_type(8))) float v8f;

#define Bq 8
#define Tq 2048
#define Dq 256
#define Qq 8
#define Kq 1024
#define Nq (Bq * Tq)            // 16384 tokens

// --------------------------------------------------------------------------
// Kernel 1: per-entry squared norms |e_k|^2 for all Q*K codebook rows,
// plus zero the loss accumulators.
// --------------------------------------------------------------------------
__global__ void vq_norms(const float* __restrict__ cb,
                         float* __restrict__ norms,
                         float* __restrict__ lossAcc) {
    const int i = blockIdx.x * blockDim.x + threadIdx.x;   // 0 .. Q*K-1
    if (blockIdx.x == 0 && threadIdx.x < Qq) lossAcc[threadIdx.x] = 0.0f;
    if (i >= Qq * Kq) return;
    const float4* p = (const float4*)(cb + (size_t)i * Dq);
    float s = 0.0f;
#pragma unroll
    for (int j = 0; j < Dq / 4; ++j) {
        float4 v = p[j];
        s += v.x * v.x + v.y * v.y + v.z * v.z + v.w * v.w;
    }
    norms[i] = s;
}

// --------------------------------------------------------------------------
// Kernel 2: one VQ layer. GEMM (scores = residual . cb^T) via
// V_WMMA_F32_16X16X4_F32 with fused argmax, gather, residual update, loss.
//   block: 256 threads = 8 waves, each wave owns 16 rows (A resident in regs)
//   codebook tiles double-buffered in LDS via GLOBAL_LOAD_ASYNC_TO_LDS_B128
//   (ASYNCcnt-tracked, zero VGPR staging), shared by all 8 waves.
// --------------------------------------------------------------------------
__launch_bounds__(256, 1)
__global__ void vq_layer(const float* in,            // residual in  [N, D]
                         float* out,                 // residual out [N, D] (may == in)
                         const float* __restrict__ cb,     // [K, D] this layer
                         const float* __restrict__ norms,  // [K]   this layer
                         float* __restrict__ outIdx,       // [N] (already offset)
                         float* __restrict__ lossAcc)      // scalar accumulator
{
    constexpr int RS   = 260;        // padded row stride (floats) -> conflict-free b64
    constexpr int TILE = 16 * RS;    // one 16-entry tile in LDS
    __shared__ float sB[2 * TILE];   // double buffer (33280 B)

    const int tid  = threadIdx.x;
    const int lane = tid & 31;
    const int w    = tid >> 5;                       // wave id 0..7
    const int rowBase = blockIdx.x * 128 + w * 16;   // 16 rows per wave
    const int half = lane >> 4;                      // 0: lanes 0-15, 1: lanes 16-31
    const int l16  = lane & 15;

    // ---- A panel resident: per-lane base = row (l16), dims (kc + 2*half + {0,1})
    const float* aBase = in + (size_t)(rowBase + l16) * Dq + 2 * half;
    v2f a[64];
#pragma unroll
    for (int j = 0; j < 64; ++j) a[j] = *(const v2f*)(aBase + 4 * j);

    // ---- async tile fetch: thread -> entry e (tid/16), float4 group g (tid%16)
    // Each thread owns 4 x b128 chunks; inst_offset moves LDS and global
    // addresses in lockstep (ISA: INST_OFFSET added to both sides).
    const int e = tid >> 4;          // 0..15  codebook entry within tile
    const int g = tid & 15;          // 0..15  float4 slot group

    // prologue: kick off tile 0 into buffer 0
    {
        const float* gp = cb + (size_t)e * Dq + g * 4;
        const unsigned lda = (unsigned)(uintptr_t)&sB[e * RS + g * 4];
        asm volatile(
            "global_load_async_to_lds_b128 %0, %1, off\n\t"
            "global_load_async_to_lds_b128 %0, %1, off offset:256\n\t"
            "global_load_async_to_lds_b128 %0, %1, off offset:512\n\t"
            "global_load_async_to_lds_b128 %0, %1, off offset:768"
            :: "v"(lda), "v"(gp) : "memory");
    }
    asm volatile("s_wait_asynccnt 0x0" ::: "memory");
    __syncthreads();

    float best[8];
    int   bidx[8];
#pragma unroll
    for (int v = 0; v < 8; ++v) { best[v] = -3.4e38f; bidx[v] = 0; }

    const int bOff = l16 * RS + 2 * half;   // B operand base inside a tile

    for (int kt = 0; kt < 64; ++kt) {
        const int cur = kt & 1;
        // kick off tile kt+1 into the alternate buffer (overlaps WMMA chain)
        if (kt < 63) {
            const float* gp = cb + (size_t)((kt + 1) * 16 + e) * Dq + g * 4;
            const unsigned lda =
                (unsigned)(uintptr_t)&sB[(cur ^ 1) * TILE + e * RS + g * 4];
            asm volatile(
                "global_load_async_to_lds_b128 %0, %1, off\n\t"
                "global_load_async_to_lds_b128 %0, %1, off offset:256\n\t"
                "global_load_async_to_lds_b128 %0, %1, off offset:512\n\t"
                "global_load_async_to_lds_b128 %0, %1, off offset:768"
                :: "v"(lda), "v"(gp) : "memory");
        }
        // 16x16 score tile: 64 chained fp32 WMMAs over D=256
        const float* bp = &sB[cur * TILE + bOff];
        v8f c = {};
#pragma unroll
        for (int j = 0; j < 64; ++j) {
            v2f bv = *(const v2f*)(bp + 4 * j);
            c = __builtin_amdgcn_wmma_f32_16x16x4_f32(
                    false, a[j], false, bv, (short)0, c, false, false);
        }
        // fused argmax epilogue: lane column n, 8 rows per lane in C layout
        const int n = kt * 16 + l16;
        const float nrm = norms[n];
#pragma unroll
        for (int v = 0; v < 8; ++v) {
            const float val = 2.0f * c[v] - nrm;
            if (val > best[v]) { best[v] = val; bidx[v] = n; }
        }
        if (kt < 63) {
            asm volatile("s_wait_asynccnt 0x0" ::: "memory");
            __syncthreads();
        }
    }

    // ---- cross-lane argmax within each 16-lane group (first-index tiebreak)
#pragma unroll
    for (int v = 0; v < 8; ++v) {
        float bv_ = best[v];
        int   bi  = bidx[v];
#pragma unroll
        for (int off = 8; off >= 1; off >>= 1) {
            const float ov = __shfl_xor(bv_, off, 32);
            const int   oi = __shfl_xor(bi,  off, 32);
            if (ov > bv_ || (ov == bv_ && oi < bi)) { bv_ = ov; bi = oi; }
        }
        bidx[v] = bi;   // lanes 0 / 16 hold winners for rows v / v+8
    }
    if (l16 == 0) {
#pragma unroll
        for (int v = 0; v < 8; ++v)
            outIdx[rowBase + half * 8 + v] = (float)bidx[v];
    }

    // ---- gather + residual update + loss (each lane: 8 dims per row)
    float lsum = 0.0f;
#pragma unroll
    for (int m = 0; m < 16; ++m) {
        const int id = __shfl(bidx[m & 7], (m >> 3) << 4, 32);
        const float4* cr = (const float4*)(cb + (size_t)id * Dq);
        const float4* xr = (const float4*)(in + (size_t)(rowBase + m) * Dq);
        float4* orow     = (float4*)(out + (size_t)(rowBase + m) * Dq);
        const int d0 = lane * 2;
#pragma unroll
        for (int j = 0; j < 2; ++j) {
            const float4 xv = xr[d0 + j];
            const float4 cv = cr[d0 + j];
            float4 r;
            r.x = xv.x - cv.x; r.y = xv.y - cv.y;
            r.z = xv.z - cv.z; r.w = xv.w - cv.w;
            orow[d0 + j] = r;
            lsum += r.x * r.x + r.y * r.y + r.z * r.z + r.w * r.w;
        }
    }
#pragma unroll
    for (int off = 16; off >= 1; off >>= 1) lsum += __shfl_xor(lsum, off, 32);
    if (lane == 0) atomicAdd(lossAcc, lsum);
}

// --------------------------------------------------------------------------
// Kernel 3: quantized_out = x - residual_final ; losses = accum / (N*D)
// --------------------------------------------------------------------------
__global__ void vq_finalize(const float* __restrict__ x,
                            const float* __restrict__ res,
                            const float* __restrict__ lossAcc,
                            float* __restrict__ out) {
    const int i = blockIdx.x * blockDim.x + threadIdx.x;
    if (i < Nq * Dq) out[i] = x[i] - res[i];
    if (i < Qq) out[(size_t)Nq * Dq + (size_t)Qq * Nq + i] =
        lossAcc[i] * (1.0f / ((float)Nq * (float)Dq));
}

// --------------------------------------------------------------------------
extern "C" void kernel_launch(void* const* d_in, const int* in_sizes, int n_in,
                              void* d_out, int out_size, void* d_ws, size_t ws_size,
                              hipStream_t stream) {
    const float* x   = (const float*)d_in[0];   // [B, T, D]
    const float* cbs = (const float*)d_in[1];   // [Q, K, D]
    float* out = (float*)d_out;

    float* res     = (float*)d_ws;              // [N, D] residual
    float* norms   = res + (size_t)Nq * Dq;     // [Q, K]
    float* lossAcc = norms + (size_t)Qq * Kq;   // [Q]

    vq_norms<<<(Qq * Kq + 255) / 256, 256, 0, stream>>>(cbs, norms, lossAcc);

    for (int l = 0; l < Qq; ++l) {
        const float* in = (l == 0) ? x : res;
        vq_layer<<<Nq / 128, 256, 0, stream>>>(
            in, res,
            cbs + (size_t)l * Kq * Dq,
            norms + (size_t)l * Kq,
            out + (size_t)Nq * Dq + (size_t)l * Nq,
            lossAcc + l);
    }

    vq_finalize<<<(Nq * Dq + 255) / 256, 256, 0, stream>>>(x, res, lossAcc, out);
}